// Branch_Cell_57990648430627
// MI455X (gfx1250) — compile-verified
//
#include <hip/hip_runtime.h>
#include <math.h>

// ---------------------------------------------------------------------------
// MI455X (gfx1250) implementation. Entirely memory-bound (~1.44 GB f32 weights,
// ~62 us at 23.3 TB/s). GEMVs use V_WMMA_F32_16X16X4_F32: A = vector chunk
// replicated across the 16 M rows, B = 4x16 weight tile loaded directly in the
// WMMA B layout, D rows all identical -> lanes 0..15 emit the 16 output
// columns from acc[0]. K/N are template constants so every B-tile address in
// the unrolled body folds into the load's 24-bit immediate offset (one pointer
// bump per 32 K-rows instead of a 64-bit VALU add per load).
// ---------------------------------------------------------------------------

typedef __attribute__((ext_vector_type(2))) float v2f;
typedef __attribute__((ext_vector_type(8))) float v8f;

#define HAS_WMMA_F32X4 __has_builtin(__builtin_amdgcn_wmma_f32_16x16x4_f32)

__device__ __forceinline__ float sigf(float x) { return 1.0f / (1.0f + expf(-x)); }

// out[n] = sum_k v[k] * W[k*N + n]   for matrix m = blockIdx.y
//   W  += m * (K*N) ; v += (m & 7) * vStride ; out += m * outStride
// 256 threads = 8 waves, 16 output columns per wave -> 128 columns per block.
template <int K, int N>
__global__ void __launch_bounds__(256) wmma_gemv_kernel(
    const float* __restrict__ Wbase,
    const float* __restrict__ vbase, int vStride,
    float* __restrict__ outbase, int outStride)
{
  __shared__ float lds_v[K];
  const int m = blockIdx.y;
  const float* W = Wbase + (size_t)m * (size_t)K * (size_t)N;
  const float* v = vbase + (size_t)(m & 7) * (size_t)vStride;
  float* out = outbase + (size_t)m * (size_t)outStride;

  // cooperative preload of the GEMV vector into LDS (K is a multiple of 256)
  for (int i = threadIdx.x; i < K; i += 256) lds_v[i] = v[i];
  __syncthreads();

  const int wave = threadIdx.x >> 5;
  const int lane = threadIdx.x & 31;
  const int col0 = (blockIdx.x * 8 + wave) << 4;  // 16 cols per wave
  const int n    = col0 + (lane & 15);
  const int rowSel = (lane >> 4) << 1;  // lanes 0-15 -> rows {0,1}; 16-31 -> {2,3}
  const int kB     = rowSel;            // matching A-operand K offset

  // Streaming base: element (rowSel, n); row r lives at base[r*N].
  const float* base = W + (size_t)rowSel * N + n;

#if HAS_WMMA_F32X4
  v8f acc = {};
#else
  float facc = 0.0f;
#endif

  for (int k = 0; k < K; k += 32) {
    if (k + 64 <= K) {  // stream-ahead prefetch (32 rows) -> global_prefetch_b8
      __builtin_prefetch(base + (size_t)(k + 32) * N, 0, 1);
      __builtin_prefetch(base + (size_t)(k + 33) * N, 0, 1);
    }
#pragma unroll
    for (int u = 0; u < 8; ++u) {
      v2f A, B;
      A.x = lds_v[k + 4 * u + kB];      // A[m][k']   = v[k']  (all M rows equal)
      A.y = lds_v[k + 4 * u + kB + 1];  // A[m][k'+1] = v[k'+1]
      B.x = base[(size_t)(k + 4 * u) * N];      // B VGPR0: tile row rowSel
      B.y = base[(size_t)(k + 4 * u) * N + N];  // B VGPR1: tile row rowSel+1
#if HAS_WMMA_F32X4
      acc = __builtin_amdgcn_wmma_f32_16x16x4_f32(
          /*neg_a=*/false, A, /*neg_b=*/false, B,
          /*c_mod=*/(short)0, acc, /*reuse_a=*/false, /*reuse_b=*/false);
#else
      facc += A.x * B.x + A.y * B.y;    // partial dot; combined across halves below
#endif
    }
  }

#if HAS_WMMA_F32X4
  if (lane < 16) out[n] = acc[0];       // D row 0 (all rows identical)
#else
  facc += __shfl_xor(facc, 16, 32);     // combine K-halves held by the lane halves
  if (lane < 16) out[n] = facc;
#endif
}

// ---- elementwise branch-cell gates -> h_m, c_m ------------------------------
__global__ void gates_kernel(const float* __restrict__ xg, const float* __restrict__ hg,
                             const float* __restrict__ c0,
                             float* __restrict__ hm, float* __restrict__ cm)
{
  const int idx = blockIdx.x * blockDim.x + threadIdx.x;  // < 8*2048
  const int S = 8 * 2048;
  float x0 = xg[idx], x1 = xg[S + idx], x2 = xg[2 * S + idx], x3 = xg[3 * S + idx];
  float g0 = hg[idx], g1 = hg[S + idx], g2 = hg[2 * S + idx], g3 = hg[3 * S + idx];
  float fg = 1.0f - fabsf((x1 - g1) / x1);
  float ig = sigf(x0 + g0);
  float gg = tanhf(x2 + g2);
  float og = sigf(x3 + g3);
  float c = fg * c0[idx] + ig * gg;
  cm[idx] = c;
  hm[idx] = og * tanhf(c);
}

// ---- layernorm over last dim (H), one block per axis ------------------------
__global__ void __launch_bounds__(256) layernorm_kernel(
    const float* __restrict__ x, const float* __restrict__ gamma,
    const float* __restrict__ beta, float* __restrict__ y, int H)
{
  __shared__ float s1[256];
  __shared__ float s2[256];
  const int a = blockIdx.x;
  x += (size_t)a * H; gamma += (size_t)a * H; beta += (size_t)a * H; y += (size_t)a * H;
  float sum = 0.0f, sq = 0.0f;
  for (int i = threadIdx.x; i < H; i += 256) { float t = x[i]; sum += t; sq += t * t; }
  s1[threadIdx.x] = sum; s2[threadIdx.x] = sq;
  __syncthreads();
  for (int off = 128; off > 0; off >>= 1) {
    if (threadIdx.x < off) {
      s1[threadIdx.x] += s1[threadIdx.x + off];
      s2[threadIdx.x] += s2[threadIdx.x + off];
    }
    __syncthreads();
  }
  const float mu  = s1[0] / (float)H;
  const float var = s2[0] / (float)H - mu * mu;  // biased, matches reference
  const float inv = rsqrtf(var + 1e-5f);
  for (int i = threadIdx.x; i < H; i += 256)
    y[i] = (x[i] - mu) * inv * gamma[i] + beta[i];
}

// ---- db gates: pre = preX + preH + bdbx + bdbh ; i_cell, c_cell -------------
__global__ void db_gates_kernel(const float* __restrict__ preX, const float* __restrict__ preH,
                                const float* __restrict__ bdbx, const float* __restrict__ bdbh,
                                const float* __restrict__ ct,
                                float* __restrict__ icell, float* __restrict__ ccell)
{
  const int idx = blockIdx.x * blockDim.x + threadIdx.x;  // < 8*1024
  const int S = 8 * 1024;
  float p0 = preX[idx] + preH[idx] + bdbx[idx] + bdbh[idx];
  float p1 = preX[S + idx] + preH[S + idx] + bdbx[S + idx] + bdbh[S + idx];
  float p2 = preX[2 * S + idx] + preH[2 * S + idx] + bdbx[2 * S + idx] + bdbh[2 * S + idx];
  float ic = sigf(p0) * ct[idx];
  float cc = sigf(p1) * tanhf(p2) + ic;
  icell[idx] = ic;
  ccell[idx] = cc;
}

// ---- axis reduction + output gate -------------------------------------------
__global__ void final_kernel(const float* __restrict__ t1, const float* __restrict__ t2,
                             const float* __restrict__ b_ilc, const float* __restrict__ b_ilh,
                             float* __restrict__ hdb)
{
  const int l = blockIdx.x * blockDim.x + threadIdx.x;  // < 1024
  float sc = 0.0f, sh = 0.0f;
#pragma unroll
  for (int a = 0; a < 8; ++a) {
    sc += t1[a * 1024 + l] + b_ilc[a * 1024 + l];
    sh += t2[a * 1024 + l] + b_ilh[a * 1024 + l];
  }
  hdb[l] = sigf(sh) * tanhf(sc);
}

extern "C" void kernel_launch(void* const* d_in, const int* in_sizes, int n_in,
                              void* d_out, int out_size, void* d_ws, size_t ws_size,
                              hipStream_t stream) {
  const float* in_   = (const float*)d_in[0];
  const float* h0    = (const float*)d_in[1];
  const float* c0    = (const float*)d_in[2];
  const float* hdb0  = (const float*)d_in[3];
  const float* Wx    = (const float*)d_in[4];   // [4,8,2048,2048]
  const float* Wh    = (const float*)d_in[5];   // [4,8,2048,2048]
  const float* Wdbx  = (const float*)d_in[6];   // [3,8,2048,1024]
  const float* Wdbh  = (const float*)d_in[7];   // [3,8,1024,1024]
  const float* bdbx  = (const float*)d_in[8];   // [3,8,1024]
  const float* bdbh  = (const float*)d_in[9];   // [3,8,1024]
  const float* W_ilc = (const float*)d_in[10];  // [8,1024,1024]
  const float* b_ilc = (const float*)d_in[11];  // [8,1024]
  const float* W_ilh = (const float*)d_in[12];  // [8,1024,1024]
  const float* b_ilh = (const float*)d_in[13];  // [8,1024]
  const float* W_ict = (const float*)d_in[14];  // [8,2048,1024]
  const float* lng   = (const float*)d_in[15];  // [2,8,2048]
  const float* lnb   = (const float*)d_in[16];  // [2,8,2048]

  float* out = (float*)d_out;
  float* hm  = out;           // [8,2048]
  float* cm  = out + 16384;   // [8,2048]
  float* hdb = out + 32768;   // [1024]

  float* ws    = (float*)d_ws;
  float* xg    = ws;            // 65536  [4,8,2048]
  float* hgb   = ws + 65536;    // 65536  [4,8,2048]
  float* ht    = ws + 131072;   // 16384  [8,2048]
  float* ctn   = ws + 147456;   // 16384  [8,2048]
  float* ct    = ws + 163840;   // 8192   [8,1024]
  float* preX  = ws + 172032;   // 24576  [3,8,1024]
  float* preH  = ws + 196608;   // 24576  [3,8,1024]
  float* icell = ws + 221184;   // 8192
  float* ccell = ws + 229376;   // 8192
  float* t1    = ws + 237568;   // 8192
  float* t2    = ws + 245760;   // 8192

  // Stage A: xg/hg GEMVs over Wx, Wh (1.07 GB stream — the dominant cost)
  wmma_gemv_kernel<2048, 2048><<<dim3(16, 32), 256, 0, stream>>>(Wx, in_, 0,    xg,  2048);
  wmma_gemv_kernel<2048, 2048><<<dim3(16, 32), 256, 0, stream>>>(Wh, h0,  2048, hgb, 2048);

  // Stage B: gates -> h_m, c_m (direct to d_out), then both layernorms
  gates_kernel<<<64, 256, 0, stream>>>(xg, hgb, c0, hm, cm);
  layernorm_kernel<<<8, 256, 0, stream>>>(hm, lng,         lnb,         ht,  2048);
  layernorm_kernel<<<8, 256, 0, stream>>>(cm, lng + 16384, lnb + 16384, ctn, 2048);

  // Stage C: c_t, preX, preH GEMVs (352 MB stream)
  wmma_gemv_kernel<2048, 1024><<<dim3(8, 8),  256, 0, stream>>>(W_ict, ctn,  2048, ct,   1024);
  wmma_gemv_kernel<2048, 1024><<<dim3(8, 24), 256, 0, stream>>>(Wdbx,  ht,   2048, preX, 1024);
  wmma_gemv_kernel<1024, 1024><<<dim3(8, 24), 256, 0, stream>>>(Wdbh,  hdb0, 0,    preH, 1024);

  db_gates_kernel<<<32, 256, 0, stream>>>(preX, preH, bdbx, bdbh, ct, icell, ccell);

  // Stage D: t1/t2 GEMVs (64 MB stream) + axis reduction
  wmma_gemv_kernel<1024, 1024><<<dim3(8, 8), 256, 0, stream>>>(W_ilc, icell, 1024, t1, 1024);
  wmma_gemv_kernel<1024, 1024><<<dim3(8, 8), 256, 0, stream>>>(W_ilh, ccell, 1024, t2, 1024);

  final_kernel<<<4, 256, 0, stream>>>(t1, t2, b_ilc, b_ilh, hdb);
}